// ParallelKerpleLog_50216757625001
// MI455X (gfx1250) — compile-verified
//
#include <hip/hip_runtime.h>

// KERPLE-log bias add: out[b,h,i,j] = x[b,h,i,j] - p[h]*log1p(a[h]*max(i-j,0))
// Memory-bound streaming kernel: 1 GiB total traffic, ~46us floor at 23.3 TB/s.

typedef float v4f __attribute__((ext_vector_type(4)));
typedef int   v4i __attribute__((ext_vector_type(4)));
typedef v4i __attribute__((address_space(1)))* g_v4i;   // global (AS1) int4*
typedef v4i __attribute__((address_space(3)))* l_v4i;   // LDS    (AS3) int4*

#define EPSV 0.01f

constexpr int SEQ           = 2048;            // sequence length (row width)
constexpr int THREADS       = 256;             // 8 waves (wave32)
constexpr int F4_PER_ROW    = SEQ / 4;         // 512 float4 per row
constexpr int F4_PER_THREAD = F4_PER_ROW / THREADS; // 2

#define HAVE_ASYNC_LDS __has_builtin(__builtin_amdgcn_global_load_async_to_lds_b128)

__global__ __launch_bounds__(THREADS)
void ParallelKerpleLog_kernel(const float* __restrict__ x,
                              const float* __restrict__ bias_p,
                              const float* __restrict__ bias_a,
                              float* __restrict__ out,
                              int H)
{
    __shared__ v4f xs[F4_PER_ROW];   // 8 KB row staging buffer

    const int row = blockIdx.x;              // row = (b*H + h)*SEQ + i
    const int i   = row & (SEQ - 1);         // query position (uniform per block)
    const int h   = (row >> 11) % H;         // head index   (uniform per block)

    const float p = fmaxf(bias_p[h], EPSV);
    const float a = fmaxf(bias_a[h], EPSV);

    const size_t base4 = (size_t)row * F4_PER_ROW;
    const v4f* gx   = (const v4f*)x + base4;
    v4f*       gout = (v4f*)out     + base4;
    const int  t    = threadIdx.x;

#if HAVE_ASYNC_LDS
    {
#pragma unroll
        for (int k = 0; k < F4_PER_THREAD; ++k) {
            const int slot = t + k * THREADS;
            // Async DMA: global -> LDS, 16B per lane, tracked by ASYNCcnt.
            // Builtin expects (int4 AS1*, int4 AS3*, imm offset, imm cpol).
            v4i* gptr = (v4i*)(v4f*)(gx + slot);   // strip const, generic bitcast
            v4i* lptr = (v4i*)(xs + slot);         // generic bitcast
            __builtin_amdgcn_global_load_async_to_lds_b128(
                (g_v4i)gptr, (l_v4i)lptr, /*offset=*/0, /*cpol=*/0);
        }
#if __has_builtin(__builtin_amdgcn_s_wait_asynccnt)
        __builtin_amdgcn_s_wait_asynccnt(0);
#else
        asm volatile("s_wait_asynccnt 0" ::: "memory");
#endif
        __syncthreads();
    }
#endif

#pragma unroll
    for (int k = 0; k < F4_PER_THREAD; ++k) {
        const int slot = t + k * THREADS;
        const int j0   = slot * 4;

#if HAVE_ASYNC_LDS
        v4f v = xs[slot];
#else
        v4f v = __builtin_nontemporal_load(gx + slot);   // th:NT, stream past L2
#endif

        v4f b;
#pragma unroll
        for (int e = 0; e < 4; ++e) {
            int d = i - (j0 + e);
            d = d > 0 ? d : 0;                           // tril: 0 above diagonal
            // log1p(a*d) = log(1 + a*d); d==0 -> log(1)=0 exactly.
            b[e] = -p * __logf(fmaf(a, (float)d, 1.0f)); // v_log_f32 path
        }
        v += b;
        __builtin_nontemporal_store(v, gout + slot);     // th:NT store
    }
}

extern "C" void kernel_launch(void* const* d_in, const int* in_sizes, int n_in,
                              void* d_out, int out_size, void* d_ws, size_t ws_size,
                              hipStream_t stream) {
    const float* x  = (const float*)d_in[0];   // [B,H,S,S] f32
    const float* bp = (const float*)d_in[1];   // [H,1,1]  f32
    const float* ba = (const float*)d_in[2];   // [H,1,1]  f32
    float*       o  = (float*)d_out;

    const int H    = in_sizes[1];              // 16
    const int rows = in_sizes[0] / SEQ;        // B*H*S = 65536 blocks

    ParallelKerpleLog_kernel<<<rows, THREADS, 0, stream>>>(x, bp, ba, o, H);
}